// FAModel_74320114090566
// MI455X (gfx1250) — compile-verified
//
#include <hip/hip_runtime.h>
#include <hip/hip_bf16.h>

typedef __attribute__((ext_vector_type(2))) float v2f;
typedef __attribute__((ext_vector_type(8))) float v8f;
typedef __attribute__((ext_vector_type(4))) unsigned int u32x4;
typedef __attribute__((ext_vector_type(8))) int i32x8;
typedef __attribute__((ext_vector_type(4))) int i32x4;

#define NN      50000
#define EE      800000
#define DIN     500
#define HH      256
#define LL      4
#define CC      7
#define ALPHA   0.5f
#define GAMMA   0.5f
#define EPSF    0.1f
#define LN_EPS  1e-5f

// ---------------------------------------------------------------------------
// Kernel 1: H0 = X @ W_init^T + b_init
// Block = 256 threads = 8 waves, owns a 32-row M stripe x all 256 cols.
// The 32x500 A-tile is DMA'd into LDS once per block by the Tensor Data
// Mover (tensor_load_to_lds, TENSORcnt), then every wave computes a 2x2 grid
// of 16x16 tiles with V_WMMA_F32_16X16X4_F32 (A from LDS, B from L2-resident
// W). TDM tensor_dim1 = remaining rows handles the ragged last block (OOB
// reads return zero); stores are row-guarded.
// f32 WMMA operand layout (ISA 7.12.2):
//   A 16x4:  lane m=lane&15, kh=lane>>4: v0=A[m][k+2kh], v1=A[m][k+2kh+1]
//   B 4x16:  lane n=lane&15, kh=lane>>4: v0=B[k+2kh][n], v1=B[k+2kh+1][n]
//   D 16x16: VGPR r: lanes 0-15 -> M=r, lanes 16-31 -> M=r+8, N=lane&15
// ---------------------------------------------------------------------------
__global__ __launch_bounds__(256) void gemm_init_kernel(
    const float* __restrict__ X, const float* __restrict__ W,
    const float* __restrict__ bias, float* __restrict__ out)
{
    __shared__ float ldsA[32 * DIN];               // 64000 B

    const int wave = threadIdx.x >> 5;
    const int lane = threadIdx.x & 31;
    const int row0 = blockIdx.x * 32;              // 1563 blocks

    if (wave == 0) {
        // --- Tensor DMA descriptor (ISA 8.3/8.4): X[row0 .. row0+31, 0:500] -> LDS
        const unsigned long long gaddr =
            (unsigned long long)(const void*)(X + (size_t)row0 * DIN);
        const unsigned lds_base = (unsigned)(unsigned long long)(void*)&ldsA[0];
        int rows_rem = NN - row0;                  // OOB rows read as zero
        if (rows_rem > 32) rows_rem = 32;

        u32x4 g0;
        g0.x = 1u;                                           // count=1 (valid), no gather
        g0.y = lds_base;                                     // lds_addr [63:32]
        g0.z = (unsigned)(gaddr & 0xffffffffu);              // global_addr lo
        g0.w = (unsigned)((gaddr >> 32) & 0x01ffffffu)       // global_addr [56:32]
             | (2u << 30);                                   // type = 2 ("image")
        i32x8 g1;
        g1[0] = (int)(2u << 16);                             // data_size = 4 bytes
        g1[1] = (int)(((unsigned)DIN & 0xffffu) << 16);      // tensor_dim0 lo16 @ [63:48]
        g1[2] = (int)(((unsigned)DIN >> 16)                  // tensor_dim0 hi16
             | (((unsigned)rows_rem & 0xffffu) << 16));      // tensor_dim1 lo16
        g1[3] = (int)(((unsigned)rows_rem >> 16)             // tensor_dim1 hi16
             | ((unsigned)DIN << 16));                       // tile_dim0 = 500
        g1[4] = 32;                                          // tile_dim1 = 32, tile_dim2 = 0
        g1[5] = (int)DIN;                                    // tensor_dim0_stride lo32
        g1[6] = 0;                                           // stride hi / dim1_stride lo
        g1[7] = 0;
        const i32x4 z4 = {0, 0, 0, 0};                       // groups 2/3 unused (2-D)
        const i32x8 z8 = {0, 0, 0, 0, 0, 0, 0, 0};
        __builtin_amdgcn_tensor_load_to_lds(g0, g1, z4, z4, z8, 0);
        __builtin_amdgcn_s_wait_tensorcnt((short)0);
    }
    __syncthreads();

    const int m  = lane & 15;
    const int kh = lane >> 4;                      // K half (0/1)
    const int nb = lane & 15;
    const int n0 = wave * 32;

    const float* la0 = ldsA + (size_t)m * DIN;          // M tile 0 (rows 0..15)
    const float* la1 = ldsA + (size_t)(m + 16) * DIN;   // M tile 1 (rows 16..31)
    const float* wr0 = W + (size_t)(n0 + nb) * DIN;     // N tile 0
    const float* wr1 = W + (size_t)(n0 + 16 + nb) * DIN;// N tile 1

    v8f acc00 = {}, acc01 = {}, acc10 = {}, acc11 = {};

    #pragma unroll 5
    for (int k = 0; k < DIN; k += 4) {
        const int ko = k + kh * 2;
        v2f a0, a1, b0, b1;
        a0.x = la0[ko]; a0.y = la0[ko + 1];
        a1.x = la1[ko]; a1.y = la1[ko + 1];
        b0.x = wr0[ko]; b0.y = wr0[ko + 1];
        b1.x = wr1[ko]; b1.y = wr1[ko + 1];
        acc00 = __builtin_amdgcn_wmma_f32_16x16x4_f32(false, a0, false, b0, (short)0, acc00, false, false);
        acc01 = __builtin_amdgcn_wmma_f32_16x16x4_f32(false, a0, false, b1, (short)0, acc01, false, false);
        acc10 = __builtin_amdgcn_wmma_f32_16x16x4_f32(false, a1, false, b0, (short)0, acc10, false, false);
        acc11 = __builtin_amdgcn_wmma_f32_16x16x4_f32(false, a1, false, b1, (short)0, acc11, false, false);
    }

    const float bia0 = bias[n0 + nb];
    const float bia1 = bias[n0 + 16 + nb];
    const int mbase = kh * 8;
    #pragma unroll
    for (int r = 0; r < 8; ++r) {
        const int rowA = row0 + mbase + r;
        if (rowA < NN) {
            const size_t o = (size_t)rowA * HH;
            out[o + n0 + nb]      = acc00[r] + bia0;
            out[o + n0 + 16 + nb] = acc01[r] + bia1;
        }
        const int rowB = rowA + 16;
        if (rowB < NN) {
            const size_t o = (size_t)rowB * HH;
            out[o + n0 + nb]      = acc10[r] + bia0;
            out[o + n0 + 16 + nb] = acc11[r] + bia1;
        }
    }
}

// ---------------------------------------------------------------------------
// Kernel 2: al[n] = x[n].att_l ; ar[n] = x[n].att_r  (one wave per node)
// ---------------------------------------------------------------------------
__global__ __launch_bounds__(256) void node_alpha_kernel(
    const float* __restrict__ x, const float* __restrict__ attl,
    const float* __restrict__ attr, float* __restrict__ al, float* __restrict__ ar)
{
    const int node = blockIdx.x * 8 + (threadIdx.x >> 5);
    if (node >= NN) return;
    const int lane = threadIdx.x & 31;
    const float4* xr = (const float4*)(x + (size_t)node * HH);
    const float4* lp = (const float4*)attl;
    const float4* rp = (const float4*)attr;
    float sl = 0.f, sr = 0.f;
    #pragma unroll
    for (int i = 0; i < 2; ++i) {
        const float4 xv = xr[lane + 32 * i];
        const float4 lv = lp[lane + 32 * i];
        const float4 rv = rp[lane + 32 * i];
        sl += xv.x * lv.x + xv.y * lv.y + xv.z * lv.z + xv.w * lv.w;
        sr += xv.x * rv.x + xv.y * rv.y + xv.z * rv.z + xv.w * rv.w;
    }
    #pragma unroll
    for (int off = 16; off > 0; off >>= 1) {
        sl += __shfl_xor(sl, off, 32);
        sr += __shfl_xor(sr, off, 32);
    }
    if (lane == 0) { al[node] = sl; ar[node] = sr; }
}

// ---------------------------------------------------------------------------
// Kernel 3: agg = EPS*x0 + ALPHA*tanh(al+ar)*x   (self-loop + residual fold)
// ---------------------------------------------------------------------------
__global__ __launch_bounds__(256) void self_init_kernel(
    const float* __restrict__ x, const float* __restrict__ x0,
    const float* __restrict__ al, const float* __restrict__ ar,
    float* __restrict__ agg)
{
    const size_t idx = (size_t)blockIdx.x * blockDim.x + threadIdx.x;
    if (idx >= (size_t)NN * (HH / 4)) return;
    const int node = (int)(idx >> 6);
    const float a = tanhf(al[node] + ar[node]) * ALPHA;
    const float4 xv  = ((const float4*)x)[idx];
    const float4 x0v = ((const float4*)x0)[idx];
    float4 o;
    o.x = EPSF * x0v.x + a * xv.x;
    o.y = EPSF * x0v.y + a * xv.y;
    o.z = EPSF * x0v.z + a * xv.z;
    o.w = EPSF * x0v.w + a * xv.w;
    ((float4*)agg)[idx] = o;
}

// ---------------------------------------------------------------------------
// Kernel 4: edge scatter. One wave per edge: read x[src] row (1 KB), scale by
// tanh(al[src]+ar[dst])*GAMMA, atomic-add into agg[dst]. agg (51 MB) is
// L2-resident on MI455X (192 MB L2) so f32 atomics stay on-chip.
// ---------------------------------------------------------------------------
__global__ __launch_bounds__(256) void edge_kernel(
    const int* __restrict__ src, const int* __restrict__ dst,
    const float* __restrict__ x, const float* __restrict__ al,
    const float* __restrict__ ar, float* __restrict__ agg)
{
    const int e = blockIdx.x * 8 + (threadIdx.x >> 5);
    if (e >= EE) return;
    const int lane = threadIdx.x & 31;
    const int s = src[e];
    const int d = dst[e];
    const float a = tanhf(al[s] + ar[d]) * GAMMA;
    const float4* xs = (const float4*)(x + (size_t)s * HH);
    float* aggd = agg + (size_t)d * HH;
    #pragma unroll
    for (int i = 0; i < 2; ++i) {
        const int c = lane + 32 * i;
        const float4 v = xs[c];
        atomicAdd(aggd + c * 4 + 0, v.x * a);
        atomicAdd(aggd + c * 4 + 1, v.y * a);
        atomicAdd(aggd + c * 4 + 2, v.z * a);
        atomicAdd(aggd + c * 4 + 3, v.w * a);
    }
}

// ---------------------------------------------------------------------------
// Kernel 5: x = LayerNorm(ReLU(x)) * g + b, in place. One wave per node.
// ---------------------------------------------------------------------------
__global__ __launch_bounds__(256) void post_ln_kernel(
    float* __restrict__ xio, const float* __restrict__ g,
    const float* __restrict__ b)
{
    const int node = blockIdx.x * 8 + (threadIdx.x >> 5);
    if (node >= NN) return;
    const int lane = threadIdx.x & 31;
    float4* row = (float4*)(xio + (size_t)node * HH);
    float4 v[2];
    float s = 0.f, s2 = 0.f;
    #pragma unroll
    for (int i = 0; i < 2; ++i) {
        float4 t = row[lane + 32 * i];
        t.x = fmaxf(t.x, 0.f); t.y = fmaxf(t.y, 0.f);
        t.z = fmaxf(t.z, 0.f); t.w = fmaxf(t.w, 0.f);
        v[i] = t;
        s  += t.x + t.y + t.z + t.w;
        s2 += t.x * t.x + t.y * t.y + t.z * t.z + t.w * t.w;
    }
    #pragma unroll
    for (int off = 16; off > 0; off >>= 1) {
        s  += __shfl_xor(s,  off, 32);
        s2 += __shfl_xor(s2, off, 32);
    }
    const float mean = s * (1.0f / HH);
    const float var  = s2 * (1.0f / HH) - mean * mean;
    const float rstd = rsqrtf(var + LN_EPS);
    #pragma unroll
    for (int i = 0; i < 2; ++i) {
        const float4 gv = ((const float4*)g)[lane + 32 * i];
        const float4 bv = ((const float4*)b)[lane + 32 * i];
        float4 o;
        o.x = (v[i].x - mean) * rstd * gv.x + bv.x;
        o.y = (v[i].y - mean) * rstd * gv.y + bv.y;
        o.z = (v[i].z - mean) * rstd * gv.z + bv.z;
        o.w = (v[i].w - mean) * rstd * gv.w + bv.w;
        row[lane + 32 * i] = o;
    }
}

// ---------------------------------------------------------------------------
// Kernel 6: emb = x @ W_last^T + b_last ; logp = log_softmax(emb).
// ---------------------------------------------------------------------------
__global__ __launch_bounds__(256) void final_kernel(
    const float* __restrict__ x, const float* __restrict__ Wl,
    const float* __restrict__ bl, float* __restrict__ emb,
    float* __restrict__ logp)
{
    const int node = blockIdx.x * 8 + (threadIdx.x >> 5);
    if (node >= NN) return;
    const int lane = threadIdx.x & 31;
    const float4* xr = (const float4*)(x + (size_t)node * HH);
    const float4 xv0 = xr[lane];
    const float4 xv1 = xr[lane + 32];

    float logits[CC];
    #pragma unroll
    for (int c = 0; c < CC; ++c) {
        const float4* wr = (const float4*)(Wl + (size_t)c * HH);
        const float4 w0 = wr[lane];
        const float4 w1 = wr[lane + 32];
        float s = xv0.x * w0.x + xv0.y * w0.y + xv0.z * w0.z + xv0.w * w0.w
                + xv1.x * w1.x + xv1.y * w1.y + xv1.z * w1.z + xv1.w * w1.w;
        #pragma unroll
        for (int off = 16; off > 0; off >>= 1) s += __shfl_xor(s, off, 32);
        logits[c] = s + bl[c];
    }
    float mx = logits[0];
    #pragma unroll
    for (int c = 1; c < CC; ++c) mx = fmaxf(mx, logits[c]);
    float se = 0.f;
    #pragma unroll
    for (int c = 0; c < CC; ++c) se += __expf(logits[c] - mx);
    const float lse = mx + __logf(se);
    if (lane == 0) {
        #pragma unroll
        for (int c = 0; c < CC; ++c) {
            emb [(size_t)node * CC + c] = logits[c];
            logp[(size_t)node * CC + c] = logits[c] - lse;
        }
    }
}

// ---------------------------------------------------------------------------
extern "C" void kernel_launch(void* const* d_in, const int* in_sizes, int n_in,
                              void* d_out, int out_size, void* d_ws, size_t ws_size,
                              hipStream_t stream) {
    (void)in_sizes; (void)n_in; (void)out_size; (void)ws_size;

    const float* x_in   = (const float*)d_in[0];
    const int*   ei     = (const int*)  d_in[1];
    const float* W_init = (const float*)d_in[2];
    const float* b_init = (const float*)d_in[3];
    const float* att_l  = (const float*)d_in[4];   // [L,H]
    const float* att_r  = (const float*)d_in[5];
    const float* ln_g   = (const float*)d_in[6];   // [L,H]
    const float* ln_b   = (const float*)d_in[7];
    const float* W_last = (const float*)d_in[8];
    const float* b_last = (const float*)d_in[9];

    const int* src = ei;        // edge_index[0]
    const int* dst = ei + EE;   // edge_index[1]

    // workspace layout (floats): x0 | bufA | bufB | al | ar  (~154 MB)
    float* ws   = (float*)d_ws;
    const size_t NH = (size_t)NN * HH;
    float* x0   = ws;
    float* bufA = ws + NH;
    float* bufB = ws + 2 * NH;
    float* al   = ws + 3 * NH;
    float* ar   = al + NN;

    const int nodeWaveBlocks = (NN + 7) / 8;           // one wave per node
    const int elemBlocks     = (int)((NH / 4 + 255) / 256);
    const int edgeBlocks     = (EE + 7) / 8;           // one wave per edge
    const int gemmBlocks     = (NN + 31) / 32;         // 32-row stripes

    // 1) x0 = X @ W_init^T + b_init   (TDM -> LDS, WMMA f32 16x16x4)
    gemm_init_kernel<<<gemmBlocks, 256, 0, stream>>>(x_in, W_init, b_init, x0);

    // 2) four FAConv layers, ping-pong bufA/bufB
    float* cur = x0;
    for (int i = 0; i < LL; ++i) {
        float* agg = (i & 1) ? bufB : bufA;
        node_alpha_kernel<<<nodeWaveBlocks, 256, 0, stream>>>(
            cur, att_l + (size_t)i * HH, att_r + (size_t)i * HH, al, ar);
        self_init_kernel<<<elemBlocks, 256, 0, stream>>>(cur, x0, al, ar, agg);
        edge_kernel<<<edgeBlocks, 256, 0, stream>>>(src, dst, cur, al, ar, agg);
        if (i != LL - 1)
            post_ln_kernel<<<nodeWaveBlocks, 256, 0, stream>>>(
                agg, ln_g + (size_t)i * HH, ln_b + (size_t)i * HH);
        cur = agg;
    }

    // 3) classifier + log_softmax; d_out = [emb (N*C) | logp (N*C)]
    float* emb  = (float*)d_out;
    float* logp = emb + (size_t)NN * CC;
    final_kernel<<<nodeWaveBlocks, 256, 0, stream>>>(cur, W_last, b_last, emb, logp);
}